// attentionWeightRecycle_37812892074642
// MI455X (gfx1250) — compile-verified
//
#include <hip/hip_runtime.h>
#include <hip/hip_bf16.h>

typedef __attribute__((ext_vector_type(2))) float v2f;
typedef __attribute__((ext_vector_type(8))) float v8f;

// Problem constants (fixed by setup_inputs)
#define NB   8
#define CCH  256
#define HH   40
#define WW2  40
#define QQ   1600     // H*W
#define QP   800      // QQ/2 (k-pair rows of Ft)
#define FHH  80
#define FWW  80
#define PPX  6400     // FH*FW
#define DFF  1024

__device__ __forceinline__ v8f wmma_f32(v2f a, v2f b, v8f c) {
    return __builtin_amdgcn_wmma_f32_16x16x4_f32(false, a, false, b, (short)0, c, false, false);
}

// generic pointer to LDS: low 32 bits are the LDS byte offset (flat aperture rule)
__device__ __forceinline__ uint32_t lds_off(const void* p) {
    return (uint32_t)(uintptr_t)p;
}
// gfx1250 async direct-to-LDS loads (ASYNCcnt-tracked, in-order completion)
__device__ __forceinline__ void async_b64(uint32_t lds, const float* g) {
    asm volatile("global_load_async_to_lds_b64 %0, %1, off" :: "v"(lds), "v"(g) : "memory");
}
__device__ __forceinline__ void async_b128(uint32_t lds, const float* g) {
    asm volatile("global_load_async_to_lds_b128 %0, %1, off" :: "v"(lds), "v"(g) : "memory");
}

// ---------------------------------------------------------------------------
// Kernel 0: pack [K][N] -> [K/2][N][2] (K-pair interleave for b64 B-fragments)
// ---------------------------------------------------------------------------
__global__ __launch_bounds__(256) void pack_pairs(const float* __restrict__ src,
                                                  float* __restrict__ dst,
                                                  int Ncols, int total /* (K/2)*N */) {
    const int idx = blockIdx.x * 256 + threadIdx.x;
    if (idx >= total) return;
    const int kp = idx / Ncols, c = idx % Ncols;
    v2f p;
    p.x = src[(size_t)(2 * kp) * Ncols + c];
    p.y = src[(size_t)(2 * kp + 1) * Ncols + c];
    ((v2f*)dst)[idx] = p;
}

// ---------------------------------------------------------------------------
// Kernel 1: 2x2 sum-pool of feature [N,C,80,80] -> Ftp [N, 800(jp), 256(c), 2]
// (K-pair-interleaved B matrix of the attention GEMM)
// ---------------------------------------------------------------------------
__global__ __launch_bounds__(256) void pool_kernel(const float* __restrict__ feat,
                                                   float* __restrict__ Ftp) {
    __shared__ float tile[40][257];
    const int b = blockIdx.x;            // N*40 blocks: one coarse row h per block
    const int n = b / HH, h = b % HH;
    const float* fbase = feat + (size_t)n * CCH * PPX;
    for (int idx = threadIdx.x; idx < CCH * WW2; idx += 256) {
        const int c = idx / WW2, w = idx % WW2;
        const float* p0 = fbase + (size_t)c * PPX + (2 * h) * FWW + 2 * w;
        tile[w][c] = p0[0] + p0[1] + p0[FWW] + p0[FWW + 1];
    }
    __syncthreads();
    // j = h*40 + w ; jp = h*20 + w/2 ; element = (c, w&1)
    float* outp = Ftp + ((size_t)n * QP + (size_t)h * 20) * 512;
    for (int idx = threadIdx.x; idx < 20 * 512; idx += 256) {
        const int jpl = idx >> 9;            // 0..19
        const int t   = idx & 511;           // c*2 + sel
        const int c = t >> 1, sel = t & 1;
        outp[(size_t)jpl * 512 + t] = tile[2 * jpl + sel][c];
    }
}

// ---------------------------------------------------------------------------
// Kernel 2: per-batch GEMM  Xa[n,i,c] = sum_j W[n,i,j] * Ft[n,j,c]
// M=1600, N=256, K=1600.  Block tile 32x256, double-buffered async-to-LDS
// staging; B held K-pair-interleaved so every fragment is one ds_load_b64.
// 8 waves: 2 M-groups x 4 N-groups; each wave 16x64 via 4 f32 WMMA accums.
// ---------------------------------------------------------------------------
__global__ __launch_bounds__(256) void attn_gemm(const float* __restrict__ Wt,
                                                 const float* __restrict__ Ftp,
                                                 float* __restrict__ Xa) {
    __shared__ float As[2][32][18];     // pad 18 -> conflict-free column reads
    __shared__ float Bsp[2][8][520];    // jp-rows of pairs; 2080B rows (16B-mult)

    const int n   = blockIdx.x / 50;        // 8 batches x 50 m-blocks
    const int m0  = (blockIdx.x % 50) * 32;
    const float* A  = Wt  + (size_t)n * QQ * QQ;
    const float* Bp = Ftp + (size_t)n * QP * 512;
    float*       D  = Xa  + (size_t)n * QQ * CCH;

    const int lane = threadIdx.x & 31, wv = threadIdx.x >> 5;
    const int half = lane >> 4, l15 = lane & 15;
    const int mg16 = (wv >> 2) * 16;        // wave's M-group inside the 32 rows
    const int n0   = (wv & 3) * 64;         // wave's 64-wide N-group

    // per-thread staging assignment (cooperative across 256 threads)
    const int ar  = threadIdx.x >> 3;             // A: 32 rows x (8 thr x 2 cols)
    const int ac  = (threadIdx.x & 7) * 2;
    const int bjr = threadIdx.x >> 5;             // B: 8 jp-rows x (32 thr x 16 f)
    const int bc0 = (threadIdx.x & 31) * 16;

    v8f acc[4] = {};

    // one K-tile (16 j's = 8 jp rows): 1x b64 (A) + 4x b128 (B) per thread
    auto issue = [&](int bufi, int kt0) {
        async_b64(lds_off(&As[bufi][ar][ac]), A + (size_t)(m0 + ar) * QQ + kt0 + ac);
        const float* gb = Bp + (size_t)((kt0 >> 1) + bjr) * 512 + bc0;
        const uint32_t lb = lds_off(&Bsp[bufi][bjr][bc0]);
        async_b128(lb,      gb);
        async_b128(lb + 16, gb + 4);
        async_b128(lb + 32, gb + 8);
        async_b128(lb + 48, gb + 12);
    };

    int buf = 0;
    issue(0, 0);
    for (int kt = 0; kt < 100; ++kt) {
        if (kt + 1 < 100) {
            issue(buf ^ 1, (kt + 1) * 16);
            // 5 outstanding from next tile; <=5 means current tile landed (in-order)
            asm volatile("s_wait_asynccnt 0x5" ::: "memory");
        } else {
            asm volatile("s_wait_asynccnt 0x0" ::: "memory");
        }
        __syncthreads();

        const float(*Asb)[18]  = As[buf];
        const float(*Bsb)[520] = Bsp[buf];
        #pragma unroll
        for (int s = 0; s < 4; ++s) {
            const int kb  = 4 * s + 2 * half;
            const int jpl = kb >> 1;
            const v2f a = *(const v2f*)&Asb[mg16 + l15][kb];
            #pragma unroll
            for (int j = 0; j < 4; ++j) {
                const v2f b = *(const v2f*)&Bsb[jpl][(n0 + 16 * j + l15) * 2];
                acc[j] = wmma_f32(a, b, acc[j]);
            }
        }
        __syncthreads();     // safe to overwrite this buffer next-next iteration
        buf ^= 1;
    }

    const int rbase = m0 + mg16 + 8 * half;
    for (int j = 0; j < 4; ++j)
        for (int g = 0; g < 8; ++g)
            D[(size_t)(rbase + g) * CCH + n0 + 16 * j + l15] = acc[j][g];
}

// ---------------------------------------------------------------------------
// Kernel 3: x1 = LN1(residual + broadcast(Xa))  -> X1 [N*6400, 256]
// ---------------------------------------------------------------------------
__global__ __launch_bounds__(256) void ln1_kernel(const float* __restrict__ feat,
                                                  const float* __restrict__ Xa,
                                                  const float* __restrict__ gw,
                                                  const float* __restrict__ gb,
                                                  float* __restrict__ X1) {
    __shared__ float tile[40][257];
    const int b  = blockIdx.x;               // N*80*2 = 1280 blocks
    const int n  = b / 160, rr = b % 160;
    const int fh = rr >> 1, fw0 = (rr & 1) * 40;
    const float* fbase = feat + (size_t)n * CCH * PPX;
    for (int idx = threadIdx.x; idx < CCH * 40; idx += 256) {
        const int c = idx / 40, w = idx % 40;
        tile[w][c] = fbase[(size_t)c * PPX + fh * FWW + fw0 + w];
    }
    __syncthreads();
    const int lane = threadIdx.x & 31, wv = threadIdx.x >> 5;
    for (int w = wv; w < 40; w += 8) {
        const int fw = fw0 + w;
        const int i  = (fh >> 1) * WW2 + (fw >> 1);
        const float* xr = Xa + ((size_t)n * QQ + i) * CCH;
        float v[8], s = 0.f, s2 = 0.f;
        for (int j = 0; j < 8; ++j) {
            const int c = lane + 32 * j;
            const float t = tile[w][c] + xr[c];
            v[j] = t; s += t; s2 += t * t;
        }
        for (int m = 16; m > 0; m >>= 1) {
            s  += __shfl_xor(s,  m, 32);
            s2 += __shfl_xor(s2, m, 32);
        }
        const float mu  = s * (1.0f / 256.0f);
        const float var = s2 * (1.0f / 256.0f) - mu * mu;
        const float rs  = rsqrtf(var + 1e-5f);
        float* orow = X1 + ((size_t)n * PPX + fh * FWW + fw) * CCH;
        for (int j = 0; j < 8; ++j) {
            const int c = lane + 32 * j;
            orow[c] = (v[j] - mu) * rs * gw[c] + gb[c];
        }
    }
}

// ---------------------------------------------------------------------------
// Kernel 4: fused MLP + LN2.  Per block: 16 rows, h kept in LDS in 128-wide
// dff chunks; x2 accumulated in WMMA registers; weights K-pair-interleaved.
// ---------------------------------------------------------------------------
__global__ __launch_bounds__(256) void mlp_kernel(const float* __restrict__ X1,
                                                  const float* __restrict__ W1p,
                                                  const float* __restrict__ B1,
                                                  const float* __restrict__ W2p,
                                                  const float* __restrict__ B2,
                                                  const float* __restrict__ g2,
                                                  const float* __restrict__ b2,
                                                  float* __restrict__ Out) {
    __shared__ float xt[16][260];
    __shared__ float ht[16][132];
    __shared__ float wk[16][260];

    const int row0 = blockIdx.x * 16;   // 3200 blocks over 51200 rows
    for (int idx = threadIdx.x; idx < 16 * 256; idx += 256) {
        const int r = idx >> 8, c = idx & 255;
        xt[r][c] = X1[(size_t)(row0 + r) * CCH + c];
    }
    __syncthreads();

    const int lane = threadIdx.x & 31, wv = threadIdx.x >> 5;
    const int half = lane >> 4, l15 = lane & 15;
    const int n0  = wv * 32;   // output-column group (GEMM2)
    const int hn0 = wv * 16;   // h-column group (GEMM1)
    v8f acc2a = {}, acc2b = {};

    for (int it = 0; it < 8; ++it) {
        const int ffb = it * 128;
        v8f acc = {};
        for (int k = 0; k < CCH; k += 4) {
            const int kb = k + 2 * half;
            const v2f a  = *(const v2f*)&xt[l15][kb];
            const v2f bf = *(const v2f*)(W1p + ((size_t)(kb >> 1) * DFF + ffb + hn0 + l15) * 2);
            acc = wmma_f32(a, bf, acc);
        }
        {
            const float bias = B1[ffb + hn0 + l15];
            const int rb = 8 * half;
            for (int g = 0; g < 8; ++g) {
                const float vh = acc[g] + bias;
                ht[rb + g][hn0 + l15] = 0.5f * vh * (1.0f + erff(vh * 0.70710678118654752f));
            }
        }
        __syncthreads();
        for (int k = 0; k < 128; k += 4) {
            const int kb = k + 2 * half;
            const v2f a = *(const v2f*)&ht[l15][kb];
            const float* w2p = W2p + ((size_t)((ffb + kb) >> 1) * CCH + n0 + l15) * 2;
            const v2f b0 = *(const v2f*)w2p;
            const v2f b1 = *(const v2f*)(w2p + 32);
            acc2a = wmma_f32(a, b0, acc2a);
            acc2b = wmma_f32(a, b1, acc2b);
        }
        __syncthreads();
    }

    {
        const int rb = 8 * half;
        const int c0 = n0 + l15, c1 = n0 + 16 + l15;
        const float bb0 = B2[c0], bb1 = B2[c1];
        for (int g = 0; g < 8; ++g) {
            wk[rb + g][c0] = acc2a[g] + bb0 + xt[rb + g][c0];
            wk[rb + g][c1] = acc2b[g] + bb1 + xt[rb + g][c1];
        }
    }
    __syncthreads();

    for (int r = wv * 2; r < wv * 2 + 2; ++r) {
        float v[8], s = 0.f, s2 = 0.f;
        for (int j = 0; j < 8; ++j) {
            const float t = wk[r][lane + 32 * j];
            v[j] = t; s += t; s2 += t * t;
        }
        for (int m = 16; m > 0; m >>= 1) {
            s  += __shfl_xor(s,  m, 32);
            s2 += __shfl_xor(s2, m, 32);
        }
        const float mu  = s * (1.0f / 256.0f);
        const float var = s2 * (1.0f / 256.0f) - mu * mu;
        const float rs  = rsqrtf(var + 1e-5f);
        float* orow = Out + (size_t)(row0 + r) * CCH;
        for (int j = 0; j < 8; ++j) {
            const int c = lane + 32 * j;
            orow[c] = (v[j] - mu) * rs * g2[c] + b2[c];
        }
    }
}

// ---------------------------------------------------------------------------
extern "C" void kernel_launch(void* const* d_in, const int* in_sizes, int n_in,
                              void* d_out, int out_size, void* d_ws, size_t ws_size,
                              hipStream_t stream) {
    (void)in_sizes; (void)n_in; (void)out_size; (void)ws_size;
    const float* weight = (const float*)d_in[0];
    const float* feature = (const float*)d_in[1];
    const float* ln1w = (const float*)d_in[2];
    const float* ln1b = (const float*)d_in[3];
    const float* w1   = (const float*)d_in[4];
    const float* b1   = (const float*)d_in[5];
    const float* w2   = (const float*)d_in[6];
    const float* b2   = (const float*)d_in[7];
    const float* ln2w = (const float*)d_in[8];
    const float* ln2b = (const float*)d_in[9];
    float* out = (float*)d_out;

    float* Ftp = (float*)d_ws;                          // [8,800,256,2]  13.1 MB
    float* Xa  = Ftp + (size_t)NB * QP * 512;           // [8,1600,256]   13.1 MB
    float* X1  = Xa  + (size_t)NB * QQ * CCH;           // [8,6400,256]   52.4 MB
    float* W1p = X1  + (size_t)NB * PPX * CCH;          // [128,1024,2]    1.0 MB
    float* W2p = W1p + (size_t)(CCH / 2) * DFF * 2;     // [512,256,2]     1.0 MB

    pack_pairs<<<(CCH / 2 * DFF + 255) / 256, 256, 0, stream>>>(w1, W1p, DFF, CCH / 2 * DFF);
    pack_pairs<<<(DFF / 2 * CCH + 255) / 256, 256, 0, stream>>>(w2, W2p, CCH, DFF / 2 * CCH);
    pool_kernel<<<NB * HH, 256, 0, stream>>>(feature, Ftp);
    attn_gemm <<<NB * 50,  256, 0, stream>>>(weight, Ftp, Xa);
    ln1_kernel<<<NB * FHH * 2, 256, 0, stream>>>(feature, Xa, ln1w, ln1b, X1);
    mlp_kernel<<<(NB * PPX) / 16, 256, 0, stream>>>(X1, W1p, b1, W2p, b2, ln2w, ln2b, out);
}